// ProjectLayer_8933531975781
// MI455X (gfx1250) — compile-verified
//
#include <hip/hip_runtime.h>
#include <hip/hip_bf16.h>
#include <stdint.h>

#define N_CAMS 5
#define Bn 4
#define Jn 15
#define Hn 128
#define Wn 240
#define CXc 80
#define CYc 80
#define CZc 20
#define NBINS (CXc * CYc * CZc)
#define HWc (Hn * Wn)
#define EPSC 1e-6f

// One thread per (batch, voxel). Camera params staged to LDS via CDNA5 async
// global->LDS loads; gathers are L2-resident (heatmaps = 36.9MB < 192MB L2).
__global__ __launch_bounds__(256) void projectlayer_cdna5_kernel(
    const float* __restrict__ hm,    // (N,B,J,H,W)
    const float* __restrict__ Rm,    // (N,B,3,3)
    const float* __restrict__ Tm,    // (N,B,3)
    const float* __restrict__ fm,    // (N,B,2)
    const float* __restrict__ cm,    // (N,B,2)
    const float* __restrict__ km,    // (N,B,3)
    const float* __restrict__ pm,    // (N,B,2)
    const float* __restrict__ whm,   // (N,B,2)
    const float* __restrict__ gcen,  // (B,3)
    const float* __restrict__ gsz,   // (3,)
    float* __restrict__ out)         // cubes (B,J,NBINS) then grids (B,NBINS,3)
{
    const int b = blockIdx.y;
    const int m = blockIdx.x * blockDim.x + threadIdx.x;  // voxel index
    const int t = threadIdx.x;

    __shared__ float sCam[N_CAMS * 23];  // per cam: R[9] T[3] f[2] c[2] k[3] p[2] wh[2]
    __shared__ float sGrid[6];           // step[3], base[3] (base includes center)

    // ---- async stage camera params into LDS (CDNA5 ASYNCcnt path) ----
    if (t < N_CAMS * 23) {
        const int n = t / 23, i = t % 23;
        const int nb = n * Bn + b;
        const float* src;
        if (i < 9)       src = Rm  + nb * 9 + i;
        else if (i < 12) src = Tm  + nb * 3 + (i - 9);
        else if (i < 14) src = fm  + nb * 2 + (i - 12);
        else if (i < 16) src = cm  + nb * 2 + (i - 14);
        else if (i < 19) src = km  + nb * 3 + (i - 16);
        else if (i < 21) src = pm  + nb * 2 + (i - 19);
        else             src = whm + nb * 2 + (i - 21);
        // Low 32 bits of a generic pointer into LDS == LDS byte offset.
        unsigned lds_addr = (unsigned)(uint64_t)(uintptr_t)&sCam[t];
        asm volatile("global_load_async_to_lds_b32 %0, %1, off"
                     :: "v"(lds_addr), "v"(src) : "memory");
    }
    if (t < 3) {
        const float s = gsz[t];
        const int nvox = (t == 0) ? CXc : ((t == 1) ? CYc : CZc);
        sGrid[t]     = s / (float)(nvox - 1);          // linspace step
        sGrid[3 + t] = -0.5f * s + gcen[b * 3 + t];    // linspace start + center
    }
    asm volatile("s_wait_asynccnt 0x0" ::: "memory");
    __syncthreads();

    // ---- voxel world coordinates ----
    const int iz  = m % CZc;
    const int iyv = (m / CZc) % CYc;
    const int ixv = m / (CZc * CYc);
    const float gx = sGrid[3] + sGrid[0] * (float)ixv;
    const float gy = sGrid[4] + sGrid[1] * (float)iyv;
    const float gz = sGrid[5] + sGrid[2] * (float)iz;

    // grids output
    {
        float* grids = out + (size_t)Bn * Jn * NBINS;
        const size_t go = ((size_t)b * NBINS + (size_t)m) * 3;
        grids[go + 0] = gx;
        grids[go + 1] = gy;
        grids[go + 2] = gz;
    }

    // ---- per-camera projection: compute once, keep in registers ----
    int   off[N_CAMS][4];
    float wgt[N_CAMS][4];
    int   base[N_CAMS];
    float den = EPSC;

#pragma unroll
    for (int n = 0; n < N_CAMS; ++n) {
        const float* cp = &sCam[n * 23];
        const float dx = gx - cp[9], dy = gy - cp[10], dz = gz - cp[11];
        const float xc = cp[0] * dx + cp[1] * dy + cp[2] * dz;
        const float yc = cp[3] * dx + cp[4] * dy + cp[5] * dz;
        const float zc = cp[6] * dx + cp[7] * dy + cp[8] * dz;
        const float invz = 1.0f / zc;
        const float u0 = xc * invz, u1 = yc * invz;
        const float r2 = u0 * u0 + u1 * u1;
        const float radial = 1.0f + r2 * (cp[16] + r2 * (cp[17] + r2 * cp[18]));
        const float tang = cp[19] * u1 + cp[20] * u0;
        const float s = radial + tang;
        const float xy0 = u0 * s + r2 * cp[20];
        const float xy1 = u1 * s + r2 * cp[19];
        float px = xy0 * cp[12] + cp[14];
        float py = xy1 * cp[13] + cp[15];
        const float wv = cp[21], hv = cp[22];
        const float bound = (px >= 0.f && py >= 0.f && px < wv && py < hv) ? 1.f : 0.f;
        den += bound;
        const float mx = fmaxf(wv, hv);
        px = fminf(fmaxf(px, -1.f), mx);
        py = fminf(fmaxf(py, -1.f), mx);
        float nx = px / (wv - 1.f) * 2.f - 1.f;
        float ny = py / (hv - 1.f) * 2.f - 1.f;
        nx = fminf(fmaxf(nx, -1.1f), 1.1f);
        ny = fminf(fmaxf(ny, -1.1f), 1.1f);
        const float ixf = (nx + 1.f) * 0.5f * (float)(Wn - 1);
        const float iyf = (ny + 1.f) * 0.5f * (float)(Hn - 1);
        const float x0f = floorf(ixf), y0f = floorf(iyf);
        const float x1f = x0f + 1.f,  y1f = y0f + 1.f;
        const float wx1 = ixf - x0f, wx0 = 1.f - wx1;
        const float wy1 = iyf - y0f, wy0 = 1.f - wy1;
        const float vx0 = (x0f >= 0.f && x0f < (float)Wn) ? 1.f : 0.f;
        const float vx1 = (x1f >= 0.f && x1f < (float)Wn) ? 1.f : 0.f;
        const float vy0 = (y0f >= 0.f && y0f < (float)Hn) ? 1.f : 0.f;
        const float vy1 = (y1f >= 0.f && y1f < (float)Hn) ? 1.f : 0.f;
        const int xi0 = (int)fminf(fmaxf(x0f, 0.f), (float)(Wn - 1));
        const int xi1 = (int)fminf(fmaxf(x1f, 0.f), (float)(Wn - 1));
        const int yi0 = (int)fminf(fmaxf(y0f, 0.f), (float)(Hn - 1));
        const int yi1 = (int)fminf(fmaxf(y1f, 0.f), (float)(Hn - 1));
        // fold corner validity AND camera bound mask into the weights
        wgt[n][0] = wx0 * wy0 * vx0 * vy0 * bound;
        wgt[n][1] = wx1 * wy0 * vx1 * vy0 * bound;
        wgt[n][2] = wx0 * wy1 * vx0 * vy1 * bound;
        wgt[n][3] = wx1 * wy1 * vx1 * vy1 * bound;
        off[n][0] = yi0 * Wn + xi0;
        off[n][1] = yi0 * Wn + xi1;
        off[n][2] = yi1 * Wn + xi0;
        off[n][3] = yi1 * Wn + xi1;
        base[n] = (n * Bn + b) * (Jn * HWc);
        // warm the gather stream for joint 0 (emits global_prefetch_b8)
        __builtin_prefetch(hm + base[n] + off[n][0], 0, 0);
    }

    const float invd = 1.0f / den;

    // ---- joints: pure gather + FMA; 20 independent L2-resident loads/iter ----
#pragma unroll
    for (int j = 0; j < Jn; ++j) {
        float acc = 0.f;
#pragma unroll
        for (int n = 0; n < N_CAMS; ++n) {
            const float* img = hm + base[n] + j * HWc;
            acc += wgt[n][0] * img[off[n][0]]
                 + wgt[n][1] * img[off[n][1]]
                 + wgt[n][2] * img[off[n][2]]
                 + wgt[n][3] * img[off[n][3]];
        }
        float r = acc * invd;
        r = fminf(fmaxf(r, 0.f), 1.f);   // also maps NaN -> 0 like reference
        out[((size_t)b * Jn + j) * NBINS + m] = r;
    }
}

extern "C" void kernel_launch(void* const* d_in, const int* in_sizes, int n_in,
                              void* d_out, int out_size, void* d_ws, size_t ws_size,
                              hipStream_t stream) {
    const float* hm   = (const float*)d_in[0];
    const float* Rm   = (const float*)d_in[1];
    const float* Tm   = (const float*)d_in[2];
    const float* fm   = (const float*)d_in[3];
    const float* cm   = (const float*)d_in[4];
    const float* km   = (const float*)d_in[5];
    const float* pm   = (const float*)d_in[6];
    const float* whm  = (const float*)d_in[7];
    const float* gcen = (const float*)d_in[8];
    const float* gsz  = (const float*)d_in[9];
    float* out = (float*)d_out;

    dim3 grid(NBINS / 256, Bn, 1);   // 500 x 4 blocks
    dim3 block(256, 1, 1);           // 8 wave32 waves per block
    projectlayer_cdna5_kernel<<<grid, block, 0, stream>>>(
        hm, Rm, Tm, fm, cm, km, pm, whm, gcen, gsz, out);
}